// LQR2_85684597555343
// MI455X (gfx1250) — compile-verified
//
#include <hip/hip_runtime.h>

// LQR dims from the reference
#define NSd  32   // states
#define NCd  16   // controls
#define NSCd 48   // NS+NC
#define Td   64   // horizon
#define NBd  512  // batch

typedef __attribute__((ext_vector_type(2))) float v2f;
typedef __attribute__((ext_vector_type(8))) float v8f;

// ---------- CDNA5 async global<->LDS path (guarded; fallback = vector copy) ----------
#if defined(__has_builtin)
#if __has_builtin(__builtin_amdgcn_global_load_async_to_lds_b128)
#define LQR_ASYNC 1
#else
#define LQR_ASYNC 0
#endif
#if LQR_ASYNC && __has_builtin(__builtin_amdgcn_global_store_async_from_lds_b128)
#define LQR_ASYNC_ST 1
#else
#define LQR_ASYNC_ST 0
#endif
#else
#define LQR_ASYNC 0
#define LQR_ASYNC_ST 0
#endif

// Exact pointee type the builtins expect: 'int __vector(4)' with AS1 (global) / AS3 (LDS).
typedef int av4i __attribute__((vector_size(4 * sizeof(int))));
typedef __attribute__((address_space(1))) av4i gbl_v4i;
typedef __attribute__((address_space(3))) av4i lds_v4i;

#if LQR_ASYNC
__device__ __forceinline__ void wait_async0() {
#if __has_builtin(__builtin_amdgcn_s_wait_asynccnt)
    __builtin_amdgcn_s_wait_asynccnt(0);
#else
    asm volatile("s_wait_asynccnt 0" ::: "memory");
#endif
}
#endif

// Stage one 48x48 f32 tile (2304 floats) from global into LDS.
// Async path: GLOBAL_LOAD_ASYNC_TO_LDS_B128 per lane (ASYNCcnt), then wait own counter.
// Caller must __syncthreads() before consuming.
__device__ __forceinline__ void stage_q_tile(const float* __restrict__ gq, float* sq, int tid) {
#if LQR_ASYNC
    for (int i = tid; i < (NSCd * NSCd) / 4; i += 128)
        __builtin_amdgcn_global_load_async_to_lds_b128(
            (gbl_v4i*)(gq + 4 * i), (lds_v4i*)(sq + 4 * i), 0, 0);
    wait_async0();
#else
    const float4* qg = (const float4*)gq;
    float4* qs = (float4*)sq;
    for (int i = tid; i < (NSCd * NSCd) / 4; i += 128) qs[i] = qg[i];
#endif
}

// ---- WMMA f32 16x16x4 tile helpers (wave32 layouts per CDNA5 ISA 7.12.2) ----
// A 16x4: lanes 0-15 hold (M=lane, K=k0,k0+1); lanes 16-31 hold (M=lane-16, K=k0+2,k0+3)
__device__ __forceinline__ v2f lda(const float* S, int ld, int m0, int k0, int lane) {
    int m = m0 + (lane & 15);
    int k = k0 + ((lane >> 4) << 1);
    v2f a; a.x = S[m * ld + k]; a.y = S[m * ld + k + 1]; return a;
}
// A-tile of S^T (element (m,k) = S[k][m])
__device__ __forceinline__ v2f ldaT(const float* S, int ld, int m0, int k0, int lane) {
    int m = m0 + (lane & 15);
    int k = k0 + ((lane >> 4) << 1);
    v2f a; a.x = S[k * ld + m]; a.y = S[(k + 1) * ld + m]; return a;
}
// B 4x16: row K striped across lanes; lanes 16-31 cover K=k0+2,k0+3
__device__ __forceinline__ v2f ldb(const float* S, int ld, int k0, int n0, int lane) {
    int n = n0 + (lane & 15);
    int k = k0 + ((lane >> 4) << 1);
    v2f b; b.x = S[k * ld + n]; b.y = S[(k + 1) * ld + n]; return b;
}
// C/D 16x16: VGPR r holds M=r (lanes 0-15) / M=r+8 (lanes 16-31), N=lane&15
__device__ __forceinline__ v8f ldc(const float* S, int ld, int m0, int n0, int lane) {
    int n  = n0 + (lane & 15);
    int mo = m0 + ((lane >> 4) << 3);
    v8f c;
#pragma unroll
    for (int r = 0; r < 8; ++r) c[r] = S[(mo + r) * ld + n];
    return c;
}
__device__ __forceinline__ void stc(float* S, int ld, int m0, int n0, int lane, v8f c, float sgn) {
    int n  = n0 + (lane & 15);
    int mo = m0 + ((lane >> 4) << 3);
#pragma unroll
    for (int r = 0; r < 8; ++r) S[(mo + r) * ld + n] = sgn * c[r];
}
__device__ __forceinline__ v8f wmma4(v2f a, v2f b, v8f c) {
    // 8 args: (neg_a, A, neg_b, B, c_mod, C, reuse_a, reuse_b)
    return __builtin_amdgcn_wmma_f32_16x16x4_f32(false, a, false, b, (short)0, c, false, false);
}

// ================= Kernel 1: nominal rollout cx + p_new = Q@xu + p =================
__global__ void __launch_bounds__(128) lqr_rollout(
    const float* __restrict__ x_init, const float* __restrict__ Q,
    const float* __restrict__ p, const float* __restrict__ A,
    float* __restrict__ cx, float* __restrict__ pnew) {
    const int b = blockIdx.x, tid = threadIdx.x;
    __shared__ __align__(16) float sQ[NSCd * NSCd];
    __shared__ float sA[NSd * NSd];
    __shared__ float sx[NSd], sxn[NSd];
    for (int i = tid; i < NSd * NSd; i += 128) sA[i] = A[i];
    if (tid < NSd) sx[tid] = x_init[b * NSd + tid];
    __syncthreads();
    for (int t = 0; t < Td; ++t) {
        const size_t bt = (size_t)b * Td + t;
        const float* pg = p + bt * NSCd;
        stage_q_tile(Q + bt * NSCd * NSCd, sQ, tid);   // async DMA into LDS
        if (tid < NSd) cx[bt * NSd + tid] = sx[tid];
        __syncthreads();
        if (tid < NSCd) {                       // p_new[j] = sum_i Q[j][i]*xu[i] + p[j], xu=[cx,0]
            float acc = pg[tid];
            const float* row = sQ + tid * NSCd;
#pragma unroll
            for (int j = 0; j < NSd; ++j) acc += row[j] * sx[j];
            pnew[bt * NSCd + tid] = acc;
        }
        if (tid < NSd) {                        // x <- x @ A^T
            float acc = 0.f;
#pragma unroll
            for (int j = 0; j < NSd; ++j) acc += sA[tid * NSd + j] * sx[j];
            sxn[tid] = acc;
        }
        __syncthreads();
        if (tid < NSd) sx[tid] = sxn[tid];
        __syncthreads();
    }
}

// ================= Kernel 2: backward Riccati scan (WMMA-heavy) =================
__global__ void __launch_bounds__(128) lqr_backward(
    const float* __restrict__ Q, const float* __restrict__ A,
    const float* __restrict__ Bm, const float* __restrict__ pnew,
    float* __restrict__ Kout, float* __restrict__ kout) {
    const int b = blockIdx.x, tid = threadIdx.x;
    const int lane = tid & 31, wave = tid >> 5;

    __shared__ __align__(16) float sQ[NSCd * NSCd];   // Qt (48x48)
    __shared__ float sF[NSd * NSCd];                  // F=[A B] (32x48)
    __shared__ float sV[NSd * NSd];                   // V  (32x32)
    __shared__ float sW[NSd * NSCd];                  // V@F (32x48)
    __shared__ __align__(16) float sK[NCd * NSd];     // Kt (16x32)
    __shared__ float sU[NCd * NSd];                   // Quu@Kt (16x32)
    __shared__ float sInv[NCd * 2 * NCd];             // [Quu | I] -> [I | Quu^-1]
    __shared__ float sVn[NSd * NSd];
    __shared__ float sv[NSd], sqt[NSCd], skt[NCd], sw2[NCd];

    for (int i = tid; i < NSd * NSCd; i += 128) {
        int r = i / NSCd, c = i % NSCd;
        sF[i] = (c < NSd) ? A[r * NSd + c] : Bm[r * NCd + (c - NSd)];
    }
    for (int i = tid; i < NSd * NSd; i += 128) sV[i] = 0.f;
    if (tid < NSd) sv[tid] = 0.f;
    __syncthreads();

    for (int t = Td - 1; t >= 0; --t) {
        const size_t bt = (size_t)b * Td + t;
        // ---- async-DMA Q tile into LDS + prefetch next Q ----
        stage_q_tile(Q + bt * NSCd * NSCd, sQ, tid);
        if (t > 0) {
            const char* qn = (const char*)(Q + (bt - 1) * NSCd * NSCd);
            __builtin_prefetch(qn + tid * 72, 0, 0);   // global_prefetch_b8 over next tile
        }
        __syncthreads();

        // ---- W = V @ F : 2x3 tiles, K=32 ----
        for (int tile = wave; tile < 6; tile += 4) {
            int mi = (tile / 3) * 16, ni = (tile % 3) * 16;
            v8f c = {};
            for (int k = 0; k < NSd; k += 4)
                c = wmma4(lda(sV, NSd, mi, k, lane), ldb(sF, NSCd, k, ni, lane), c);
            stc(sW, NSCd, mi, ni, lane, c, 1.f);
        }
        __syncthreads();

        // ---- Qt = Q + F^T @ W : 3x3 tiles, K=32 ----
        for (int tile = wave; tile < 9; tile += 4) {
            int mi = (tile / 3) * 16, ni = (tile % 3) * 16;
            v8f c = ldc(sQ, NSCd, mi, ni, lane);
            for (int k = 0; k < NSd; k += 4)
                c = wmma4(ldaT(sF, NSCd, mi, k, lane), ldb(sW, NSCd, k, ni, lane), c);
            stc(sQ, NSCd, mi, ni, lane, c, 1.f);
        }
        __syncthreads();

        // ---- qt = p_new + v@F ; augmented [Quu | I] ----
        if (tid < NSCd) {
            float acc = pnew[bt * NSCd + tid];
            for (int i = 0; i < NSd; ++i) acc += sv[i] * sF[i * NSCd + tid];
            sqt[tid] = acc;
        }
        for (int i = tid; i < NCd * 2 * NCd; i += 128) {
            int r = i >> 5, c = i & 31;
            sInv[i] = (c < NCd) ? sQ[(NSd + r) * NSCd + (NSd + c)]
                                : ((c - NCd == r) ? 1.f : 0.f);
        }
        __syncthreads();

        // ---- Gauss-Jordan: Quu is SPD (Q SPD + V PSD) so inv == pinv ----
        for (int k = 0; k < NCd; ++k) {
            float piv = sInv[k * 32 + k];
            __syncthreads();
            if (tid < 32) sInv[k * 32 + tid] *= (1.f / piv);
            __syncthreads();
            float fac[4];
            int c = tid & 31;
#pragma unroll
            for (int i = 0; i < 4; ++i) {
                int r = (tid >> 5) + 4 * i;
                fac[i] = (r == k) ? 0.f : sInv[r * 32 + k];
            }
            __syncthreads();
#pragma unroll
            for (int i = 0; i < 4; ++i) {
                int r = (tid >> 5) + 4 * i;
                sInv[r * 32 + c] -= fac[i] * sInv[k * 32 + c];
            }
            __syncthreads();
        }

        // ---- Kt = -Quu^-1 @ Qux (1x2 tiles, K=16); kt = -Quu^-1 @ qu ----
        if (wave < 2) {
            int ni = wave * 16;
            v8f c = {};
            for (int k = 0; k < NCd; k += 4)
                c = wmma4(lda(sInv, 2 * NCd, 0, NCd + k, lane),
                          ldb(sQ, NSCd, NSd + k, ni, lane), c);
            stc(sK, NSd, 0, ni, lane, c, -1.f);
        } else if (wave == 2 && lane < NCd) {
            float acc = 0.f;
            for (int j = 0; j < NCd; ++j) acc += sInv[lane * 32 + NCd + j] * sqt[NSd + j];
            skt[lane] = -acc;
        }
        __syncthreads();

        // ---- U = Quu @ Kt (1x2 tiles, K=16); w2 = Quu @ kt ----
        if (wave < 2) {
            int ni = wave * 16;
            v8f c = {};
            for (int k = 0; k < NCd; k += 4)
                c = wmma4(lda(sQ + NSd * NSCd + NSd, NSCd, 0, k, lane),
                          ldb(sK, NSd, k, ni, lane), c);
            stc(sU, NSd, 0, ni, lane, c, 1.f);
        } else if (wave == 2 && lane < NCd) {
            float acc = 0.f;
            for (int j = 0; j < NCd; ++j) acc += sQ[(NSd + lane) * NSCd + NSd + j] * skt[j];
            sw2[lane] = acc;
        }
        __syncthreads();

        // ---- Vn = Qxx + Qxu K + K^T Qux + K^T U : 2x2 tiles, C chained ----
        {
            int mi = (wave >> 1) * 16, ni = (wave & 1) * 16;
            v8f c = ldc(sQ, NSCd, mi, ni, lane);                       // Qxx tile
            for (int k = 0; k < NCd; k += 4)
                c = wmma4(lda(sQ, NSCd, mi, NSd + k, lane), ldb(sK, NSd, k, ni, lane), c);
            for (int k = 0; k < NCd; k += 4)
                c = wmma4(ldaT(sK, NSd, mi, k, lane), ldb(sQ + NSd * NSCd, NSCd, k, ni, lane), c);
            for (int k = 0; k < NCd; k += 4)
                c = wmma4(ldaT(sK, NSd, mi, k, lane), ldb(sU, NSd, k, ni, lane), c);
            stc(sVn, NSd, mi, ni, lane, c, 1.f);
        }
        __syncthreads();

        // ---- vn = qx + Qxu kt + K^T (qu + Quu kt); commit V; spill K,k ----
        if (tid < NSd) {
            float acc = sqt[tid];
            for (int j = 0; j < NCd; ++j) {
                acc += sQ[tid * NSCd + NSd + j] * skt[j];
                acc += sK[j * NSd + tid] * (sqt[NSd + j] + sw2[j]);
            }
            sv[tid] = acc;
        }
        for (int i = tid; i < NSd * NSd; i += 128) sV[i] = sVn[i];
#if LQR_ASYNC_ST
        {   // async LDS -> global spill of Kt (exactly one b128 per lane)
            float* gK = Kout + bt * NCd * NSd;
            __builtin_amdgcn_global_store_async_from_lds_b128(
                (gbl_v4i*)(gK + 4 * tid), (lds_v4i*)((float*)sK + 4 * tid), 0, 0);
            wait_async0();   // LDS sK reused next iteration; keep it simple
        }
#else
        for (int i = tid; i < NCd * NSd; i += 128) Kout[bt * NCd * NSd + i] = sK[i];
#endif
        if (tid < NCd) kout[bt * NCd + tid] = skt[tid];
        __syncthreads();
    }
}

// ================= Kernel 3: forward rollout + cost =================
__global__ void __launch_bounds__(128) lqr_forward(
    const float* __restrict__ x_init, const float* __restrict__ Q,
    const float* __restrict__ p, const float* __restrict__ A,
    const float* __restrict__ Bm, const float* __restrict__ cx,
    const float* __restrict__ Kin, const float* __restrict__ kin,
    float* __restrict__ xout, float* __restrict__ uout, float* __restrict__ cost) {
    const int b = blockIdx.x, tid = threadIdx.x;
    __shared__ float sA[NSd * NSd], sB[NSd * NCd];
    __shared__ float sx[NSd], su[NCd], sxu[NSCd], sxn[NSd];
    __shared__ float sred[128];
    for (int i = tid; i < NSd * NSd; i += 128) sA[i] = A[i];
    for (int i = tid; i < NSd * NCd; i += 128) sB[i] = Bm[i];
    if (tid < NSd) sx[tid] = x_init[b * NSd + tid];
    float costAcc = 0.f;
    __syncthreads();
    for (int t = 0; t < Td; ++t) {
        const size_t bt = (size_t)b * Td + t;
        const float* Kg  = Kin + bt * NCd * NSd;
        const float* cxg = cx + bt * NSd;
        if (tid < NCd) {                         // u = K (x - cx) + k  (cu == 0)
            float acc = kin[bt * NCd + tid];
            for (int j = 0; j < NSd; ++j) acc += Kg[tid * NSd + j] * (sx[j] - cxg[j]);
            su[tid] = acc;
        }
        __syncthreads();
        if (tid < NSCd) sxu[tid] = (tid < NSd) ? sx[tid] : su[tid - NSd];
        if (tid < NSd) xout[bt * NSd + tid] = sx[tid];
        if (tid < NCd) uout[bt * NCd + tid] = su[tid];
        __syncthreads();
        const float* Qg = Q + bt * NSCd * NSCd;
        const float* pg = p + bt * NSCd;
        if (tid < NSCd) {                        // cost += 0.5 xu'Q xu + xu.p
            float rd = 0.f;
            for (int c = 0; c < NSCd; ++c) rd += Qg[tid * NSCd + c] * sxu[c];
            costAcc += sxu[tid] * (0.5f * rd + pg[tid]);
        }
        if (tid < NSd) {                         // x <- A x + B u
            float acc = 0.f;
            for (int j = 0; j < NSd; ++j) acc += sA[tid * NSd + j] * sx[j];
            for (int j = 0; j < NCd; ++j) acc += sB[tid * NCd + j] * su[j];
            sxn[tid] = acc;
        }
        __syncthreads();
        if (tid < NSd) sx[tid] = sxn[tid];
        __syncthreads();
    }
    sred[tid] = costAcc;
    __syncthreads();
    for (int s = 64; s > 0; s >>= 1) {
        if (tid < s) sred[tid] += sred[tid + s];
        __syncthreads();
    }
    if (tid == 0) cost[b] = sred[0];
}

extern "C" void kernel_launch(void* const* d_in, const int* in_sizes, int n_in,
                              void* d_out, int out_size, void* d_ws, size_t ws_size,
                              hipStream_t stream) {
    const float* x_init = (const float*)d_in[0];
    const float* Q      = (const float*)d_in[1];
    const float* p      = (const float*)d_in[2];
    const float* A      = (const float*)d_in[3];
    const float* Bm     = (const float*)d_in[4];
    // d_in[5] = dt (unused by reference math)

    float* ws      = (float*)d_ws;
    float* ws_pnew = ws;                                     // NB*T*NSC
    float* ws_cx   = ws_pnew + (size_t)NBd * Td * NSCd;      // NB*T*NS
    float* ws_K    = ws_cx   + (size_t)NBd * Td * NSd;       // NB*T*NC*NS
    float* ws_k    = ws_K    + (size_t)NBd * Td * NCd * NSd; // NB*T*NC

    float* xout = (float*)d_out;
    float* uout = xout + (size_t)NBd * Td * NSd;
    float* cost = uout + (size_t)NBd * Td * NCd;

    lqr_rollout <<<NBd, 128, 0, stream>>>(x_init, Q, p, A, ws_cx, ws_pnew);
    lqr_backward<<<NBd, 128, 0, stream>>>(Q, A, Bm, ws_pnew, ws_K, ws_k);
    lqr_forward <<<NBd, 128, 0, stream>>>(x_init, Q, p, A, Bm, ws_cx, ws_K, ws_k,
                                          xout, uout, cost);
}